// ImpalaLoss_58351425683661
// MI455X (gfx1250) — compile-verified
//
#include <hip/hip_runtime.h>

// Problem constants (match reference)
#define BB 512
#define TT 1024
#define AA 32
#define ROWS (BB * TT)          // 524288
#define P1_BLOCK 256
#define P1_GRID (ROWS / P1_BLOCK)   // 2048 entropy partials
#define DISCOUNT 0.99f
#define ENT_COEF 0.05f

typedef float v2f __attribute__((ext_vector_type(2)));
typedef float v8f __attribute__((ext_vector_type(8)));

// ---------------------------------------------------------------------------
// Row statistics for log-softmax over 32 contiguous floats, fully in VGPRs.
// m = max, s = sum exp(x-m), tt = sum exp(x-m)*(x-m)
// ---------------------------------------------------------------------------
__device__ __forceinline__ void row_stats(const float* __restrict__ row,
                                          float& m, float& s, float& tt) {
    float x[AA];
    const float4* r4 = reinterpret_cast<const float4*>(row);
#pragma unroll
    for (int i = 0; i < 8; ++i) {
        float4 q = r4[i];
        x[4 * i + 0] = q.x; x[4 * i + 1] = q.y;
        x[4 * i + 2] = q.z; x[4 * i + 3] = q.w;
    }
    m = x[0];
#pragma unroll
    for (int i = 1; i < AA; ++i) m = fmaxf(m, x[i]);
    s = 0.0f; tt = 0.0f;
#pragma unroll
    for (int i = 0; i < AA; ++i) {
        float d = x[i] - m;
        float e = __expf(d);
        s += e;
        tt = fmaf(e, d, tt);
    }
}

// ---------------------------------------------------------------------------
// Phase 1: streaming pass over both logits tensors (134 MB -> memory bound).
// One lane per (b,t) row. Emits time-major CE and min(rho,1), entropy partials.
// ---------------------------------------------------------------------------
__global__ __launch_bounds__(P1_BLOCK) void impala_phase1(
    const float* __restrict__ beh_logits,   // [B,T,A]
    const float* __restrict__ tgt_logits,   // [B,T,A]
    const int*   __restrict__ action,       // [B,T]
    float* __restrict__ CE,                 // [T,B] time-major
    float* __restrict__ CMIN,               // [T,B] time-major
    float* __restrict__ ENTP)               // [P1_GRID]
{
    const int gid = blockIdx.x * P1_BLOCK + threadIdx.x;   // row id = b*T + t
    const long base = (long)gid * AA;

    // target (learner) row: need lse, entropy, logit[action]
    float mt, st, ttt;
    row_stats(tgt_logits + base, mt, st, ttt);
    const float lst = __logf(st);
    const float lse_t = mt + lst;
    const float ent_row = ttt / st - lst;       // sum p*logp for this row

    // behavior (actor) row: need only lse
    float mb, sb, tb;
    row_stats(beh_logits + base, mb, sb, tb);
    const float lse_b = mb + __logf(sb);

    const int a = action[gid];
    const float t_alp = tgt_logits[base + a] - lse_t;
    const float b_alp = beh_logits[base + a] - lse_b;

    const float log_rho = t_alp - b_alp;
    const float rho = __expf(log_rho);
    const float cmin = fminf(rho, 1.0f);        // = clipped_rho = cs = clipped_pg_rho

    // time-major scatter (dword stores; all L2-resident, tiny vs logits traffic)
    const int b = gid >> 10;          // / T
    const int t = gid & (TT - 1);     // % T
    const int idx2 = t * BB + b;
    CE[idx2] = -t_alp;
    CMIN[idx2] = cmin;

    // block reduction of entropy partials (deterministic tree)
    float v = ent_row;
    v += __shfl_xor(v, 16, 32);
    v += __shfl_xor(v, 8, 32);
    v += __shfl_xor(v, 4, 32);
    v += __shfl_xor(v, 2, 32);
    v += __shfl_xor(v, 1, 32);
    __shared__ float wpart[P1_BLOCK / 32];
    const int lane = threadIdx.x & 31;
    const int wid = threadIdx.x >> 5;
    if (lane == 0) wpart[wid] = v;
    __syncthreads();
    if (threadIdx.x == 0) {
        float tsum = 0.0f;
#pragma unroll
        for (int i = 0; i < P1_BLOCK / 32; ++i) tsum += wpart[i];
        ENTP[blockIdx.x] = tsum;
    }
}

// ---------------------------------------------------------------------------
// Phase 2: V-trace backward scan, one lane per sequence b (512 lanes).
// Single backward pass produces vs-v (acc), pg_adv, and accumulates losses.
// CE/CMIN loads are coalesced (time-major); values/reward/done are L2-hot.
// ---------------------------------------------------------------------------
__global__ __launch_bounds__(256) void impala_phase2(
    const float* __restrict__ CE,            // [T,B]
    const float* __restrict__ CMIN,          // [T,B]
    const float* __restrict__ values,        // [B,T]
    const float* __restrict__ reward,        // [B,T]
    const int*   __restrict__ done,          // [B,T]
    const float* __restrict__ next_values,   // [B,T]
    float* __restrict__ PGP,                 // [512]
    float* __restrict__ BASEP)               // [512]
{
    const int b = blockIdx.x * blockDim.x + threadIdx.x;   // 0..511
    const float boot = next_values[(long)b * TT + (TT - 1)];
    float vnext = boot;     // values_{t+1} (bootstrap at t=T-1)
    float vsnext = boot;    // vs_{t+1}     (bootstrap at t=T-1)
    float acc = 0.0f;       // vs_minus_v carried backward
    float pg = 0.0f, base = 0.0f;

    for (int t = TT - 1; t >= 0; --t) {
        const int i2 = t * BB + b;          // time-major, coalesced
        const long i1 = (long)b * TT + t;   // row-major inputs
        const float c  = CMIN[i2];
        const float ce = CE[i2];
        const float v  = values[i1];
        const float r  = reward[i1];
        const float disc = done[i1] ? 0.0f : DISCOUNT;

        const float q = r - v;
        const float delta = c * fmaf(disc, vnext, q);
        acc = fmaf(disc * c, acc, delta);            // vs_minus_v[t]
        const float pgadv = c * fmaf(disc, vsnext, q);
        pg = fmaf(ce, pgadv, pg);
        base = fmaf(0.5f * acc, acc, base);          // 0.5*(vs-values)^2
        vsnext = acc + v;                            // vs[t]
        vnext = v;
    }
    PGP[b] = pg;
    BASEP[b] = base;
}

// ---------------------------------------------------------------------------
// Phase 3: deterministic final reduction with V_WMMA_F32_16X16X4_F32.
// A = ones(16x4)  =>  D[m][n] = sum_k B[k][n] + C  (column sums).
// Each wmma folds 64 fp32 partials; then 4x shfl_xor folds 16 column sums.
// One wave, EXEC all ones (WMMA requirement).
// ---------------------------------------------------------------------------
__device__ __forceinline__ float wmma_sum(const float* __restrict__ p, int n,
                                          int lane) {
    v8f c = {0.f, 0.f, 0.f, 0.f, 0.f, 0.f, 0.f, 0.f};
    v2f ones; ones.x = 1.0f; ones.y = 1.0f;
    for (int i = 0; i < n; i += 64) {
        v2f bm;
        bm.x = p[i + lane];
        bm.y = p[i + 32 + lane];
        c = __builtin_amdgcn_wmma_f32_16x16x4_f32(
            /*neg_a=*/false, ones, /*neg_b=*/false, bm,
            /*c_mod=*/(short)0, c, /*reuse_a=*/false, /*reuse_b=*/false);
    }
    // every lane's c[0] holds colsum(lane & 15); fold the 16 columns
    float v = c[0];
    v += __shfl_xor(v, 8, 32);
    v += __shfl_xor(v, 4, 32);
    v += __shfl_xor(v, 2, 32);
    v += __shfl_xor(v, 1, 32);
    return v;   // lanes 0..15 (and 16..31) all hold the total
}

__global__ __launch_bounds__(32) void impala_final(
    const float* __restrict__ ENTP,
    const float* __restrict__ PGP,
    const float* __restrict__ BASEP,
    float* __restrict__ out)
{
    const int lane = threadIdx.x;
    const float ent  = wmma_sum(ENTP, P1_GRID, lane);
    const float pg   = wmma_sum(PGP, BB, lane);
    const float base = wmma_sum(BASEP, BB, lane);
    if (lane == 0) {
        out[0] = pg + base + ent * ENT_COEF;   // total (BASELINE_COEF = 1)
        out[1] = pg;
        out[2] = base;
        out[3] = ent;
    }
}

// ---------------------------------------------------------------------------
// Launch
// ---------------------------------------------------------------------------
extern "C" void kernel_launch(void* const* d_in, const int* in_sizes, int n_in,
                              void* d_out, int out_size, void* d_ws, size_t ws_size,
                              hipStream_t stream) {
    (void)in_sizes; (void)n_in; (void)out_size; (void)ws_size;

    const float* beh   = (const float*)d_in[0];   // behavior_policy_logits [B,T,A]
    const float* tgt   = (const float*)d_in[1];   // target_policy_logits  [B,T,A]
    const int*   act   = (const int*)d_in[2];     // action [B,T] (int32 in JAX default)
    const float* vals  = (const float*)d_in[3];   // values [B,T]
    const float* nvals = (const float*)d_in[4];   // next_values [B,T]
    const float* rew   = (const float*)d_in[5];   // reward [B,T]
    const int*   done  = (const int*)d_in[6];     // done [B,T]
    float* out = (float*)d_out;

    // Workspace layout (floats): CE[T*B], CMIN[T*B], ENTP[2048], PGP[512], BASEP[512]
    float* ws    = (float*)d_ws;
    float* CE    = ws;
    float* CMIN  = CE + ROWS;
    float* ENTP  = CMIN + ROWS;
    float* PGP   = ENTP + P1_GRID;
    float* BASEP = PGP + BB;

    impala_phase1<<<P1_GRID, P1_BLOCK, 0, stream>>>(beh, tgt, act, CE, CMIN, ENTP);
    impala_phase2<<<BB / 256, 256, 0, stream>>>(CE, CMIN, vals, rew, done, nvals,
                                                PGP, BASEP);
    impala_final<<<1, 32, 0, stream>>>(ENTP, PGP, BASEP, out);
}